// PathATTN_11836929868042
// MI455X (gfx1250) — compile-verified
//
#include <hip/hip_runtime.h>
#include <hip/hip_bf16.h>

typedef __attribute__((ext_vector_type(2))) float v2f;
typedef __attribute__((ext_vector_type(4))) float v4f;
typedef __attribute__((ext_vector_type(8))) float v8f;

#define CDIM 512
#define TT   256
#define TS   256
#define NEGINF (-1e9f)

// ---------------------------------------------------------------------------
// Kernel 1: collapse the linear-linear score MLP:
//   v[c] = sum_d w2[d] * w1[d,c] ;  c0 = sum_d w2[d]*b1[d] + b2
// ---------------------------------------------------------------------------
__global__ __launch_bounds__(256) void collapse_weights(
    const float* __restrict__ w1_w, const float* __restrict__ w1_b,
    const float* __restrict__ w2_w, const float* __restrict__ w2_b,
    float* __restrict__ vout) {
  const int c = blockIdx.x * 256 + threadIdx.x;   // 0..511
  float acc = 0.f;
  for (int d = 0; d < CDIM; ++d)
    acc = fmaf(w2_w[d], w1_w[d * CDIM + c], acc); // w2 reads are wave-uniform
  vout[c] = acc;
  if (blockIdx.x == 0 && threadIdx.x == 0) {
    float b = w2_b[0];
    for (int d = 0; d < CDIM; ++d) b = fmaf(w2_w[d], w1_b[d], b);
    vout[CDIM] = b;
  }
}

// ---------------------------------------------------------------------------
// Kernel 2: fused wait-k attention for one (b,t) tile per block.
//   scores via V_WMMA_F32_16X16X4_F32, softmax in LDS, weighted sum over s.
// ---------------------------------------------------------------------------
__global__ __launch_bounds__(256) void waitk_attn(
    const float* __restrict__ x, const float* __restrict__ vb,
    const int* __restrict__ waitk_p, float* __restrict__ out) {
  __shared__ __align__(16) float s_v[CDIM];
  __shared__ __align__(16) float s_alpha[TS];
  __shared__ float s_red[8];

  const int tid  = threadIdx.x;
  const int t    = blockIdx.x;
  const int b    = blockIdx.y;
  const int wave = tid >> 5;
  const int lane = tid & 31;
  const int half = lane >> 4;     // which half-wave (selects K pair / M offset)
  const int rowl = lane & 15;     // row within 16-row group

  const int   waitk = waitk_p[0];
  const float c0    = vb[CDIM];
  const float* xt   = x + (size_t)(b * TT + t) * (TS * CDIM);

  // stage v into LDS; init all scores to masked
  s_v[tid]       = vb[tid];
  s_v[tid + 256] = vb[tid + 256];
  s_alpha[tid]   = NEGINF;
  __syncthreads();

  const int S_eff = min(TS, t + waitk);       // rows with nonzero probability
  const int G     = (S_eff + 15) >> 4;        // 16-row groups to score

  // ---- Stage A: scores alpha[s] = x[s,:]·v + c0 via f32 WMMA ----
  // k-slot->channel mapping per WMMA pair: lane half h covers channels
  // {cb+4h, cb+4h+1} (WMMA#1) and {cb+4h+2, cb+4h+3} (WMMA#2), so each lane's
  // A operand for both WMMAs is a single contiguous b128 load. B is v
  // broadcast across N => every D column equals the row's dot product.
  for (int g = wave; g < G; g += 8) {
    const int s0 = g << 4;
    const float* arow = xt + (size_t)(s0 + rowl) * CDIM + 4 * half;
    const float* vrow = s_v + 4 * half;
    v8f d = {};
#pragma unroll 4
    for (int cb = 0; cb < CDIM; cb += 8) {
      v4f xa = *(const v4f*)(arow + cb);
      v4f vv = *(const v4f*)(vrow + cb);
      v2f a1 = {xa.x, xa.y}, a2 = {xa.z, xa.w};
      v2f b1 = {vv.x, vv.y}, b2 = {vv.z, vv.w};
      d = __builtin_amdgcn_wmma_f32_16x16x4_f32(false, a1, false, b1,
                                                (short)0, d, false, false);
      d = __builtin_amdgcn_wmma_f32_16x16x4_f32(false, a2, false, b2,
                                                (short)0, d, false, false);
    }
    // D VGPR j: lanes 0-15 hold M=j, lanes 16-31 hold M=j+8 (all N identical)
    if (rowl == 0) {
#pragma unroll
      for (int j = 0; j < 8; ++j) {
        const int s = s0 + half * 8 + j;
        float val = d[j] + c0;
        if (s >= t + waitk) val = NEGINF;
        s_alpha[s] = val;
      }
    }
  }
  __syncthreads();

  // ---- softmax over Ts in LDS ----
  const float sc = s_alpha[tid];
  float m = sc;
#pragma unroll
  for (int off = 16; off; off >>= 1) m = fmaxf(m, __shfl_xor(m, off, 32));
  if (lane == 0) s_red[wave] = m;
  __syncthreads();
  m = s_red[0];
#pragma unroll
  for (int w = 1; w < 8; ++w) m = fmaxf(m, s_red[w]);

  const float p = __expf(sc - m);   // masked rows: exp(-1e9 - m) == 0
  float sum = p;
#pragma unroll
  for (int off = 16; off; off >>= 1) sum += __shfl_xor(sum, off, 32);
  __syncthreads();                  // all reads of s_red done before rewrite
  if (lane == 0) s_red[wave] = sum;
  __syncthreads();
  float tot = 0.f;
#pragma unroll
  for (int w = 0; w < 8; ++w) tot += s_red[w];
  const float pn = p / tot;
  s_alpha[tid] = pn;                // each thread owns its own slot
  __syncthreads();

  // ---- Stage B: out[b,t,c] = sum_s p[s] * x[s,c], skip masked tail ----
  const int S4 = (S_eff + 3) & ~3;  // <= 256; padding rows have p == 0
  const float* xc = xt + 2 * tid;
  v2f acc = {0.f, 0.f};
  for (int s = 0; s < S4; s += 4) {
    v4f pv = *(const v4f*)(s_alpha + s);
    v2f x0 = *(const v2f*)(xc + (size_t)(s + 0) * CDIM);
    v2f x1 = *(const v2f*)(xc + (size_t)(s + 1) * CDIM);
    v2f x2 = *(const v2f*)(xc + (size_t)(s + 2) * CDIM);
    v2f x3 = *(const v2f*)(xc + (size_t)(s + 3) * CDIM);
    acc += pv.x * x0 + pv.y * x1 + pv.z * x2 + pv.w * x3;
  }
  *(v2f*)(out + (size_t)(b * TT + t) * CDIM + 2 * tid) = acc;
}

// ---------------------------------------------------------------------------
extern "C" void kernel_launch(void* const* d_in, const int* in_sizes, int n_in,
                              void* d_out, int out_size, void* d_ws, size_t ws_size,
                              hipStream_t stream) {
  const float* x     = (const float*)d_in[0];
  const float* w1_w  = (const float*)d_in[1];
  const float* w1_b  = (const float*)d_in[2];
  const float* w2_w  = (const float*)d_in[3];
  const float* w2_b  = (const float*)d_in[4];
  const int*   waitk = (const int*)d_in[5];
  float* outp = (float*)d_out;
  float* vbuf = (float*)d_ws;   // 513 floats: v[512] + c0

  collapse_weights<<<dim3(2), 256, 0, stream>>>(w1_w, w1_b, w2_w, w2_b, vbuf);
  waitk_attn<<<dim3(TT, 4), 256, 0, stream>>>(x, vbuf, waitk, outp);
}